// BatchLpsmap_35957466202386
// MI455X (gfx1250) — compile-verified
//
#include <hip/hip_runtime.h>
#include <stdint.h>

// Problem constants (from reference)
#define NV       16384
#define CC       512
#define KK       64
#define NBATCH   64
#define MAX_IT   25
#define NBISECT  25
#define FBUDGET  8.0f
// RHO == 1.0 -> folded into the math below.

typedef float v2f  __attribute__((ext_vector_type(2)));
typedef float v8f  __attribute__((ext_vector_type(8)));
typedef unsigned int u32x4 __attribute__((ext_vector_type(4)));
typedef int i32x4 __attribute__((ext_vector_type(4)));
typedef int i32x8 __attribute__((ext_vector_type(8)));

__device__ __forceinline__ float clip01(float x) {
  return fminf(fmaxf(x, 0.0f), 1.0f);
}

// Sum over K=64 of f(m) for the 16 constraint rows owned by this wave.
// Rows are placed along the WMMA N dimension: A = ones(16x4), B = data(4x16).
// D[m,n] = sum_k B[k,n] = rowsum(n)  -> every D VGPR at lane l holds
// rowsum(l%16), i.e. exactly the lanes that own that row. No shuffles needed.
template <typename F>
__device__ __forceinline__ float wmma_rowsum(F f) {
  v2f ones; ones[0] = 1.0f; ones[1] = 1.0f;
  v8f acc = {};
#pragma unroll
  for (int c = 0; c < 16; ++c) {
    v2f bb;
    bb[0] = f(2 * c);
    bb[1] = f(2 * c + 1);
    acc = __builtin_amdgcn_wmma_f32_16x16x4_f32(false, ones, false, bb,
                                                (short)0, acc, false, false);
  }
  return acc[0];
}

__global__ __launch_bounds__(1024, 1) void lpsmap_admm_kernel(
    const float* __restrict__ scores, const int* __restrict__ cidx,
    float* __restrict__ out) {
  // LDS: [0,NV) = t (scatter accumulator) then u (in place); [NV,2NV) = scores
  __shared__ float smem[2 * NV];
  float* s_tu = smem;
  float* s_sc = smem + NV;

  const int tid  = threadIdx.x;          // 0..1023
  const int wave = tid >> 5;             // 0..31
  const int lane = tid & 31;
  const int half = lane >> 4;            // lane pair selector (K phase 0/2)
  const int row  = wave * 16 + (lane & 15);   // constraint row owned (0..511)
  const int b    = blockIdx.x;

  // ---- TDM: async DMA this batch's 64KB score slice into LDS (wave 0) ----
  if (wave == 0) {
    uint64_t ga = (uint64_t)(uintptr_t)(scores + (size_t)b * NV);
    u32x4 g0;
    g0[0] = 1u;                                   // count=1 (valid), no gather
    g0[1] = (unsigned)(NV * 4);                   // lds byte offset of s_sc
    g0[2] = (unsigned)ga;                         // global_addr[31:0]
    g0[3] = ((unsigned)(ga >> 32) & 0x01FFFFFFu)  // global_addr[56:32]
            | (2u << 30);                         // type=2 ("image")
    i32x8 g1;
    g1[0] = 0x00020000;        // workgroup_mask=0, data_size=2 (4 bytes)
    g1[1] = 0x40000000;        // tensor_dim0[15:0]=0x4000 in bits[31:16]
    g1[2] = 0x00010000;        // tensor_dim0 hi=0; tensor_dim1 lo16=1
    g1[3] = 0x40000000;        // tensor_dim1 hi=0; tile_dim0=0x4000 (16384)
    g1[4] = 0;                 // tile_dim1=0 (unused), tile_dim2=0
    g1[5] = NV;                // tensor_dim0_stride lo32
    g1[6] = 0;
    g1[7] = 0;
    i32x4 gz = {0, 0, 0, 0};
#if defined(__clang_major__) && (__clang_major__ >= 23)
    i32x8 gz8 = {0, 0, 0, 0, 0, 0, 0, 0};
    __builtin_amdgcn_tensor_load_to_lds(g0, g1, gz, gz, gz8, 0);
#else
    __builtin_amdgcn_tensor_load_to_lds(g0, g1, gz, gz, 0);
#endif
  }

  // ---- Load this lane's 32 constraint indices, packed 2x16-bit (idx < 16384)
  // Lane l<16 owns row elements k in {4c,4c+1}; lane l>=16 owns {4c+2,4c+3}.
  unsigned idp[16];
#pragma unroll
  for (int c = 0; c < 16; ++c) {
    int k0 = 4 * c + 2 * half;
    unsigned i0 = (unsigned)cidx[row * KK + k0];
    unsigned i1 = (unsigned)cidx[row * KK + k0 + 1];
    idp[c] = (i0 & 0xFFFFu) | (i1 << 16);
  }

  // ---- Degree pass: deg[n] = #appearances of n in idx ----
#pragma unroll
  for (int j = 0; j < 16; ++j) s_tu[tid + 1024 * j] = 0.0f;
  if (wave == 0) __builtin_amdgcn_s_wait_tensorcnt(0);
  __syncthreads();
#pragma unroll
  for (int c = 0; c < 16; ++c) {
    atomicAdd(&s_tu[idp[c] & 0xFFFFu], 1.0f);
    atomicAdd(&s_tu[idp[c] >> 16], 1.0f);
  }
  __syncthreads();
  unsigned degmask = 0;       // bit j: deg(tid + 1024*j) > 0
  float invR[16];             // 1 / (RHO * deg_safe)
#pragma unroll
  for (int j = 0; j < 16; ++j) {
    float d = s_tu[tid + 1024 * j];
    degmask |= (d > 0.0f ? 1u : 0u) << j;
    invR[j] = 1.0f / fmaxf(d, 1.0f);
  }
  __syncthreads();

  // ---- ADMM state in registers ----
  float zR[32], lamR[32];
#pragma unroll
  for (int m = 0; m < 32; ++m) { zR[m] = 0.0f; lamR[m] = 0.0f; }

  for (int it = 0; it <= MAX_IT; ++it) {
    // zero scatter accumulator
#pragma unroll
    for (int j = 0; j < 16; ++j) s_tu[tid + 1024 * j] = 0.0f;
    __syncthreads();
    if (it > 0) {  // it==0: z=lam=0 -> msg=0, t stays 0 (matches reference)
#pragma unroll
      for (int c = 0; c < 16; ++c) {
        atomicAdd(&s_tu[idp[c] & 0xFFFFu], zR[2 * c]     - lamR[2 * c]);
        atomicAdd(&s_tu[idp[c] >> 16],     zR[2 * c + 1] - lamR[2 * c + 1]);
      }
    }
    __syncthreads();
    // u update (in place: each thread owns its 16 slots)
#pragma unroll
    for (int j = 0; j < 16; ++j) {
      int n = tid + 1024 * j;
      float sc = s_sc[n];
      float uu = clip01((sc + s_tu[n]) * invR[j]);
      if (!((degmask >> j) & 1u)) uu = (sc > 0.0f) ? 1.0f : 0.0f;
      s_tu[n] = uu;
      if (it == MAX_IT) out[(size_t)b * NV + n] = uu;
    }
    if (it == MAX_IT) break;
    __syncthreads();
    // gather Au = u[idx]
    float au[32];
#pragma unroll
    for (int c = 0; c < 16; ++c) {
      au[2 * c]     = s_tu[idp[c] & 0xFFFFu];
      au[2 * c + 1] = s_tu[idp[c] >> 16];
    }
    __syncthreads();  // protect next iteration's zeroing of s_tu

    // ---- capped-simplex projection of a = Au + lam (per row, lane-pair) ----
    float s0 = wmma_rowsum([&](int m) { return clip01(au[m] + lamR[m]); });
    bool feasible = (s0 <= FBUDGET);
    float mn = 1e30f, mx = -1e30f;
#pragma unroll
    for (int m = 0; m < 32; ++m) {
      float a = au[m] + lamR[m];
      mn = fminf(mn, a);
      mx = fmaxf(mx, a);
    }
    mn = fminf(mn, __shfl_xor(mn, 16, 32));
    mx = fmaxf(mx, __shfl_xor(mx, 16, 32));
    float lo = mn - 1.0f, hi = mx;
    for (int bi = 0; bi < NBISECT; ++bi) {
      float mid = 0.5f * (lo + hi);
      float s = wmma_rowsum([&](int m) { return clip01(au[m] + lamR[m] - mid); });
      bool gt = (s > FBUDGET);
      lo = gt ? mid : lo;
      hi = gt ? hi : mid;
    }
    float tau0 = 0.5f * (lo + hi);
    float nact = wmma_rowsum([&](int m) {
      float d = au[m] + lamR[m] - tau0;
      return (d > 0.0f && d < 1.0f) ? 1.0f : 0.0f;
    });
    float g = wmma_rowsum([&](int m) { return clip01(au[m] + lamR[m] - tau0); })
              - FBUDGET;
    float tau = tau0 + g / fmaxf(nact, 1.0f);
#pragma unroll
    for (int m = 0; m < 32; ++m) {
      float a  = au[m] + lamR[m];
      float zn = feasible ? clip01(a) : clip01(a - tau);
      zR[m]    = zn;
      lamR[m]  = lamR[m] + (au[m] - zn);   // lam += RHO*(Au - z_new)
    }
  }
}

extern "C" void kernel_launch(void* const* d_in, const int* in_sizes, int n_in,
                              void* d_out, int out_size, void* d_ws, size_t ws_size,
                              hipStream_t stream) {
  const float* scores = (const float*)d_in[0];   // (64, 16384) f32
  const int*   cidx   = (const int*)d_in[1];     // (512, 64) i32
  float*       out    = (float*)d_out;           // (64, 16384) f32
  (void)in_sizes; (void)n_in; (void)out_size; (void)d_ws; (void)ws_size;
  hipLaunchKernelGGL(lpsmap_admm_kernel, dim3(NBATCH), dim3(1024), 0, stream,
                     scores, cidx, out);
}